// NewsEncoder_59639915872560
// MI455X (gfx1250) — compile-verified
//
#include <hip/hip_runtime.h>
#include <hip/hip_bf16.h>

typedef float v2f __attribute__((ext_vector_type(2)));
typedef float v8f __attribute__((ext_vector_type(8)));

#define H_   20
#define DH_  20
#define D_   300
#define AH_  200
#define S_   30
#define B_   512
#define HD_  (H_*DH_)      // 400
#define EQ_  (D_*H_)       // 6000
#define CB_  (S_*D_)       // 9000 floats of c per batch elem
#define XQB_ (S_*EQ_)      // 180000 floats of xq per batch elem
#define XVB_ (S_*HD_)      // 12000 floats of xv / h per batch elem
#define CHUNK_B 64
#define NCHUNK  (B_/CHUNK_B)

// workspace layout (in floats)
#define OFF_C   ((size_t)0)
#define OFF_XV  (OFF_C  + (size_t)B_*CB_)        // 4,608,000
#define OFF_H   (OFF_XV + (size_t)B_*XVB_)       // +6,144,000
#define OFF_XQ  (OFF_H  + (size_t)B_*XVB_)       // +6,144,000  (chunk buffer: 11,520,000)

// ---------------------------------------------------------------------------
// 1) c = emb[x] + sinusoidal PE   (one element per thread)
// ---------------------------------------------------------------------------
__global__ void embed_pe_kernel(const int* __restrict__ x,
                                const float* __restrict__ emb,
                                float* __restrict__ c) {
    int e   = blockIdx.x * blockDim.x + threadIdx.x;   // 0 .. B*S*D-1 (exact)
    int row = e / D_;                                  // b*S + s
    int d   = e - row * D_;
    int s   = row % S_;
    int tok = x[row];
    int de  = d & ~1;
    float div = __expf((float)de * (-9.210340371976184f / (float)D_));
    float ang = (float)s * div;
    float pe  = (d & 1) ? __cosf(ang) : __sinf(ang);
    c[e] = emb[(size_t)tok * D_ + d] + pe;
}

// ---------------------------------------------------------------------------
// 2) C[M,N] = A[M,K] @ W[N,K]^T + bias[N]  via V_WMMA_F32_16X16X4_F32
//    32x32 macro-tile per wave (2x2 WMMA tiles, fragments reused twice).
//    Requires M%32==0, K%4==0; ragged N handled by wave-uniform clamp.
// ---------------------------------------------------------------------------
__global__ void gemm_nt_wmma(const float* __restrict__ A,
                             const float* __restrict__ W,
                             const float* __restrict__ bias,
                             float* __restrict__ C,
                             int M, int N, int K) {
    int wave   = (blockIdx.x * blockDim.x + threadIdx.x) >> 5;
    int ntiles = N >> 4;
    int npt    = (ntiles + 1) >> 1;                    // n macro-tiles
    int mpt    = M >> 5;                               // m macro-tiles
    if (wave >= mpt * npt) return;                     // wave-uniform
    int mp = wave / npt;
    int np = wave - mp * npt;
    int lane = threadIdx.x & 31;
    int hf   = lane >> 4;                              // half-wave id
    int lm   = lane & 15;
    int nt0  = np * 2, nt1 = nt0 + 1;
    bool n1ok = (nt1 < ntiles);                        // wave-uniform

    const float* a0 = A + (size_t)(mp * 32 + lm) * K;          // A rows m..m+15
    const float* a1 = a0 + (size_t)16 * K;                     // A rows m+16..m+31
    const float* b0 = W + (size_t)(nt0 * 16 + lm) * K;         // W rows = B^T cols
    const float* b1 = n1ok ? (W + (size_t)(nt1 * 16 + lm) * K) : b0;

    v8f acc00 = {0.f,0.f,0.f,0.f,0.f,0.f,0.f,0.f};
    v8f acc01 = acc00, acc10 = acc00, acc11 = acc00;
    for (int k = 0; k < K; k += 4) {
        int ko = k + 2 * hf;                           // lane's K pair
        if ((k & 63) == 0) {                           // prefetch 256B ahead
            __builtin_prefetch(a0 + ko + 64, 0, 0);
            __builtin_prefetch(a1 + ko + 64, 0, 0);
            __builtin_prefetch(b0 + ko + 64, 0, 0);
            __builtin_prefetch(b1 + ko + 64, 0, 0);
        }
        v2f av0 = *(const v2f*)(a0 + ko);
        v2f av1 = *(const v2f*)(a1 + ko);
        v2f bv0 = *(const v2f*)(b0 + ko);
        v2f bv1 = *(const v2f*)(b1 + ko);
        acc00 = __builtin_amdgcn_wmma_f32_16x16x4_f32(false, av0, false, bv0, (short)0, acc00, false, false);
        acc01 = __builtin_amdgcn_wmma_f32_16x16x4_f32(false, av0, false, bv1, (short)0, acc01, false, false);
        acc10 = __builtin_amdgcn_wmma_f32_16x16x4_f32(false, av1, false, bv0, (short)0, acc10, false, false);
        acc11 = __builtin_amdgcn_wmma_f32_16x16x4_f32(false, av1, false, bv1, (short)0, acc11, false, false);
    }
    int   col0 = nt0 * 16 + lm;
    int   col1 = col0 + 16;
    float bc0  = bias[col0];
    float bc1  = n1ok ? bias[col1] : 0.f;
    float* r0 = C + (size_t)(mp * 32 + 8 * hf) * N;    // D rows: +r (tile row pair)
    float* r1 = r0 + (size_t)16 * N;
#pragma unroll
    for (int r = 0; r < 8; ++r) {
        r0[(size_t)r * N + col0] = acc00[r] + bc0;
        r1[(size_t)r * N + col0] = acc10[r] + bc0;
        if (n1ok) {
            r0[(size_t)r * N + col1] = acc01[r] + bc1;
            r1[(size_t)r * N + col1] = acc11[r] + bc1;
        }
    }
}

// ---------------------------------------------------------------------------
// 3) per (b,h): w = Xq_slice(30x300) @ C_b(300x30); softmax over query dim;
//    h = w @ Xv_slice(30x20).  4 waves, 2x2 WMMA tiles padded 30->32.
//    Xv slice staged via gfx1250 async-to-LDS, overlapped with the WMMA loop.
// ---------------------------------------------------------------------------
__global__ void attn_wmma(const float* __restrict__ xq_c,   // chunk xq [CHUNK_B*XQB_]
                          const float* __restrict__ cbuf,   // full c
                          const float* __restrict__ xv,     // full xv
                          float* __restrict__ hbuf,         // full h
                          int b0) {
    __shared__ float sW[32 * 33];
    __shared__ float sXv[S_ * DH_];                    // 600

    int lb  = blockIdx.x / H_;
    int h   = blockIdx.x - lb * H_;
    int bg  = b0 + lb;
    int tid = threadIdx.x;                             // 0..127

    // async-stage xv slice into LDS (ASYNCcnt path); drained after WMMA loop
    for (int t = tid; t < S_ * DH_; t += 128) {
        unsigned int ldsa = (unsigned int)(unsigned long long)(&sXv[t]);
        const float* gp = xv + (size_t)bg * XVB_ + h * (S_ * DH_) + t;
        asm volatile("global_load_async_to_lds_b32 %0, %1, off"
                     :: "v"(ldsa), "v"(gp) : "memory");
    }

    int wv   = tid >> 5;
    int lane = tid & 31;
    int hf   = lane >> 4;
    int lm   = lane & 15;
    int m0   = (wv >> 1) * 16;
    int n0   = (wv & 1) * 16;
    int arow = m0 + lm;                                // query row i
    int bcol = n0 + lm;                                // key col j
    bool aok = arow < S_;
    bool bok = bcol < S_;
    const float* aptr = xq_c + (size_t)lb * XQB_ + (size_t)h * CB_ + (size_t)arow * D_;
    const float* cb   = cbuf + (size_t)bg * CB_;       // [300,30] reinterpretation

    v8f acc = {0.f,0.f,0.f,0.f,0.f,0.f,0.f,0.f};
    for (int k = 0; k < D_; k += 4) {
        int ko = k + 2 * hf;
        v2f av, bv;
        av.x = aok ? aptr[ko]     : 0.f;
        av.y = aok ? aptr[ko + 1] : 0.f;
        bv.x = bok ? cb[(size_t)ko * S_ + bcol]       : 0.f;
        bv.y = bok ? cb[(size_t)(ko + 1) * S_ + bcol] : 0.f;
        acc = __builtin_amdgcn_wmma_f32_16x16x4_f32(
                  false, av, false, bv, (short)0, acc, false, false);
    }

    // drain this wave's async LDS writes before publishing at the barrier
    asm volatile("s_wait_asynccnt 0x0" ::: "memory");

#pragma unroll
    for (int r = 0; r < 8; ++r)
        sW[(m0 + r + 8 * hf) * 33 + (n0 + lm)] = acc[r];
    __syncthreads();

    // softmax over query dim i (axis=-2) for each key column j
    if (tid < S_) {
        int j = tid;
        float mx = -1e30f;
        for (int i = 0; i < S_; ++i) mx = fmaxf(mx, sW[i * 33 + j]);
        float sum = 0.f;
        for (int i = 0; i < S_; ++i) {
            float e = __expf(sW[i * 33 + j] - mx);
            sW[i * 33 + j] = e;
            sum += e;
        }
        float inv = 1.f / sum;
        for (int i = 0; i < S_; ++i) sW[i * 33 + j] *= inv;
    }
    __syncthreads();

    // h[i,dh] = sum_j w[i,j] * xv_slice[j,dh]   (30x30x20, VALU)
    for (int t = tid; t < S_ * DH_; t += 128) {
        int i  = t / DH_;
        int dh = t - i * DH_;
        float a2 = 0.f;
        for (int j = 0; j < S_; ++j) a2 += sW[i * 33 + j] * sXv[j * DH_ + dh];
        hbuf[(size_t)bg * XVB_ + h * (S_ * DH_) + t] = a2;
    }
}

// ---------------------------------------------------------------------------
// 4) per b: a = tanh(h_r @ WV^T + bV); score = a@wq2^T+bq2; softmax over s;
//    out = sum_s p[s] * h_r[s,:]    (deterministic tree reductions, no atomics)
// ---------------------------------------------------------------------------
__global__ void pool_kernel(const float* __restrict__ hbuf,
                            const float* __restrict__ WVm,
                            const float* __restrict__ bV,
                            const float* __restrict__ wq2,
                            const float* __restrict__ bq2,
                            float* __restrict__ out) {
    __shared__ float sH[XVB_];                         // 12000 floats = 48KB
    __shared__ float sRed[256];
    __shared__ float sScore[S_];
    __shared__ float sP[S_];
    int b = blockIdx.x, tid = threadIdx.x;

    for (int t = tid; t < XVB_; t += 256)
        sH[t] = hbuf[(size_t)b * XVB_ + t];
    __syncthreads();

    for (int s = 0; s < S_; ++s) {
        float val = 0.f;
        if (tid < AH_) {
            const float* wrow = WVm + (size_t)tid * HD_;
            const float* hrow = sH + s * HD_;
            float dot = 0.f;
            for (int k = 0; k < HD_; ++k) dot += hrow[k] * wrow[k];
            val = tanhf(dot + bV[tid]) * wq2[tid];
        }
        sRed[tid] = val;
        __syncthreads();
        for (int off = 128; off > 0; off >>= 1) {
            if (tid < off) sRed[tid] += sRed[tid + off];
            __syncthreads();
        }
        if (tid == 0) sScore[s] = sRed[0] + bq2[0];
        __syncthreads();
    }
    if (tid == 0) {
        float mx = -1e30f;
        for (int s = 0; s < S_; ++s) mx = fmaxf(mx, sScore[s]);
        float sum = 0.f;
        for (int s = 0; s < S_; ++s) { float e = __expf(sScore[s] - mx); sP[s] = e; sum += e; }
        float inv = 1.f / sum;
        for (int s = 0; s < S_; ++s) sP[s] *= inv;
    }
    __syncthreads();
    for (int k = tid; k < HD_; k += 256) {
        float z = 0.f;
        for (int s = 0; s < S_; ++s) z += sP[s] * sH[s * HD_ + k];
        out[(size_t)b * HD_ + k] = z;
    }
}

// ---------------------------------------------------------------------------
extern "C" void kernel_launch(void* const* d_in, const int* in_sizes, int n_in,
                              void* d_out, int out_size, void* d_ws, size_t ws_size,
                              hipStream_t stream) {
    const int*   x   = (const int*)  d_in[0];
    const float* emb = (const float*)d_in[1];
    const float* Wq  = (const float*)d_in[2];
    const float* bq  = (const float*)d_in[3];
    const float* Wv  = (const float*)d_in[4];
    const float* bv  = (const float*)d_in[5];
    const float* WVm = (const float*)d_in[6];
    const float* bV  = (const float*)d_in[7];
    const float* wq2 = (const float*)d_in[8];
    const float* bq2 = (const float*)d_in[9];
    float* ws = (float*)d_ws;
    float* c  = ws + OFF_C;
    float* xv = ws + OFF_XV;
    float* hb = ws + OFF_H;
    float* xq = ws + OFF_XQ;
    float* out = (float*)d_out;

    // 1) embedding + positional encoding
    embed_pe_kernel<<<(B_ * S_ * D_) / 256, 256, 0, stream>>>(x, emb, c);

    // 2) xv = c @ Wv^T + bv   (M=15360, N=400, K=300)
    {
        int M = B_ * S_, N = HD_, K = D_;
        int ntiles = N / 16, npt = (ntiles + 1) / 2, mpt = M / 32;
        int waves = mpt * npt;
        gemm_nt_wmma<<<(waves + 7) / 8, 256, 0, stream>>>(c, Wv, bv, xv, M, N, K);
    }

    // 3) per 64-batch chunk: xq GEMM (L2-resident, 46MB) then fused attention
    for (int ch = 0; ch < NCHUNK; ++ch) {
        int M = CHUNK_B * S_, N = EQ_, K = D_;
        int ntiles = N / 16, npt = (ntiles + 1) / 2, mpt = M / 32;
        int waves = mpt * npt;
        gemm_nt_wmma<<<(waves + 7) / 8, 256, 0, stream>>>(
            c + (size_t)ch * CHUNK_B * CB_, Wq, bq, xq, M, N, K);
        attn_wmma<<<CHUNK_B * H_, 128, 0, stream>>>(xq, c, xv, hb, ch * CHUNK_B);
    }

    // 4) additive attention pooling -> [B, 400]
    pool_kernel<<<B_, 256, 0, stream>>>(hb, WVm, bV, wq2, bq2, out);
}